// SparseConv_24489903522143
// MI455X (gfx1250) — compile-verified
//
#include <hip/hip_runtime.h>
#include <hip/hip_bf16.h>

typedef __attribute__((ext_vector_type(2))) float v2f;
typedef __attribute__((ext_vector_type(8))) float v8f;

#define Bq 32
#define Eq 4096

// One wave (32 threads) per block handles 16 consecutive elements of E.
// CT = number of 16-wide N tiles covering c (ceil(c/16)).
template <int CT>
__global__ __launch_bounds__(32)
void sconv_layer(const float* __restrict__ space,   // [B][E][4]
                 const int*   __restrict__ nbr,     // [B][E][16]
                 const float* __restrict__ alin,    // [B][E][fin]
                 float*       __restrict__ alout,   // [B][E][c]
                 const float* __restrict__ wfW,     // [64][16]
                 const float* __restrict__ wfB,     // [16]
                 const float* __restrict__ preW,    // [fin][c]
                 const float* __restrict__ preB,    // [c]
                 const float* __restrict__ outW,    // [16c][c]
                 const float* __restrict__ outB,    // [c]
                 int fin, int c)
{
    __shared__ float sGA[8][16][40];   // 20KB: gathered features, 8 elems (unscaled)
    __shared__ float sPre[16][640];    // 40KB: flattened pre results (16c <= 640)
    __shared__ float sW[16][16];       // softmax weights per element/neighbor
    __shared__ int   sN[16][16];       // neighbor indices

    // delta (16x64 = 4KB) lives in the not-yet-used sPre region during wfac
    float (*sDelta)[64] = (float (*)[64])(&sPre[0][0]);

    const int lane = threadIdx.x;          // 0..31
    const int b    = blockIdx.x >> 8;      // E/16 = 256 tiles per batch
    const int e0   = (blockIdx.x & 255) * 16;

    // ---- stage neighbor indices ----
    for (int i = lane; i < 256; i += 32) {
        int e = i >> 4, n = i & 15;
        sN[e][n] = nbr[(b * Eq + e0 + e) * 16 + n];
    }
    __syncthreads();

    // ---- build delta = gathered_space - space ----
    for (int i = lane; i < 1024; i += 32) {
        int e = i >> 6;
        int rem = i & 63;
        int n = rem >> 2, d = rem & 3;
        float sv = space[(b * Eq + e0 + e) * 4 + d];
        float nv = space[(b * Eq + sN[e][n]) * 4 + d];
        sDelta[e][rem] = nv - sv;
    }
    __syncthreads();

    const int row = lane & 15;                 // N-column (B/C/D) or M-row (A)
    const int khi = (lane < 16) ? 0 : 2;       // K sub-offset for 32-bit A/B frags
    const int mhi = (lane < 16) ? 0 : 8;       // M offset for C/D rows

    // ---- wfac: [16 elems x 64] @ [64 x 16], f32 WMMA, 16 K-steps ----
    {
        v8f acc = {};
        #pragma unroll
        for (int k = 0; k < 16; ++k) {
            v2f a, bb;
            a.x  = sDelta[row][4 * k + khi];
            a.y  = sDelta[row][4 * k + khi + 1];
            bb.x = wfW[(4 * k + khi) * 16 + row];
            bb.y = wfW[(4 * k + khi + 1) * 16 + row];
            acc = __builtin_amdgcn_wmma_f32_16x16x4_f32(false, a, false, bb,
                                                        (short)0, acc, false, false);
        }
        float bias = wfB[row];
        #pragma unroll
        for (int r = 0; r < 8; ++r)
            sW[mhi + r][row] = acc[r] + bias;   // logits: sW[elem][nbr]
    }
    __syncthreads();

    // ---- row softmax over the 16 neighbors (lanes 0..15, one row each) ----
    if (lane < 16) {
        float mx = sW[lane][0];
        for (int n = 1; n < 16; ++n) mx = fmaxf(mx, sW[lane][n]);
        float s = 0.f;
        float ex[16];
        for (int n = 0; n < 16; ++n) { ex[n] = __expf(sW[lane][n] - mx); s += ex[n]; }
        float inv = 1.f / s;
        for (int n = 0; n < 16; ++n) sW[lane][n] = ex[n] * inv;
    }
    __syncthreads();

    const int finp   = (fin + 3) & ~3;         // pad K to multiple of 4
    const int ksteps = finp >> 2;

    const float*   srcbase = alin + (size_t)b * Eq * fin;   // batch base (SGPRs)
    const unsigned sgaBase = (unsigned)(uintptr_t)(&sGA[0][0][0]);

    // ---- pre: [16 nbr x fin] @ [fin x c], register-blocked over 8 elements ----
    for (int sub = 0; sub < 2; ++sub) {
        const int eb = sub * 8;

        // async gather: 128 rows (8 elems x 16 nbrs) of fin floats, global -> LDS
        // one row per wave-half so all 32 lanes issue copies
        for (int rp = 0; rp < 64; ++rp) {
            int r  = rp * 2 + (lane >> 4);         // 0..127
            int e8 = r >> 4, n = r & 15;
            int gi = sN[eb + e8][n];
            unsigned lbase = sgaBase + (unsigned)(r * 40) * 4u;
            unsigned gbase = (unsigned)(gi * fin) * 4u;
            for (int j = (lane & 15); j < fin; j += 16) {
                unsigned laddr = lbase + 4u * (unsigned)j;
                unsigned gaddr = gbase + 4u * (unsigned)j;
                asm volatile("global_load_async_to_lds_b32 %0, %1, %2"
                             :
                             : "v"(laddr), "v"(gaddr), "s"(srcbase)
                             : "memory");
            }
        }
        // zero-fill K padding columns so fragment reads need no guards
        if (finp > fin) {
            for (int r = lane; r < 128; r += 32) {
                int e8 = r >> 4, n = r & 15;
                for (int j = fin; j < finp; ++j) sGA[e8][n][j] = 0.f;
            }
        }
        asm volatile("s_wait_asynccnt 0x0" ::: "memory");
        __syncthreads();

        // softmax weight per register-blocked element (A row = neighbor = `row`)
        float wv8[8];
        #pragma unroll
        for (int e8 = 0; e8 < 8; ++e8) wv8[e8] = sW[eb + e8][row];

        #pragma unroll
        for (int nt = 0; nt < CT; ++nt) {
            const int colj = nt * 16 + row;
            v8f zero = {};
            v8f acc[8];
            #pragma unroll
            for (int e8 = 0; e8 < 8; ++e8) acc[e8] = zero;

            for (int k = 0; k < ksteps; ++k) {
                int k0 = 4 * k + khi, k1 = k0 + 1;
                v2f bb;   // one B fragment, reused by 8 independent WMMAs
                bb.x = (colj < c && k0 < fin) ? preW[k0 * c + colj] : 0.f;
                bb.y = (colj < c && k1 < fin) ? preW[k1 * c + colj] : 0.f;
                #pragma unroll
                for (int e8 = 0; e8 < 8; ++e8) {
                    v2f a;   // scale folded into the A read (co-executes with WMMA)
                    a.x = sGA[e8][row][k0] * wv8[e8];
                    a.y = sGA[e8][row][k1] * wv8[e8];
                    acc[e8] = __builtin_amdgcn_wmma_f32_16x16x4_f32(false, a, false, bb,
                                                                    (short)0, acc[e8],
                                                                    false, false);
                }
            }
            if (colj < c) {
                float bias = preB[colj];
                #pragma unroll
                for (int e8 = 0; e8 < 8; ++e8) {
                    #pragma unroll
                    for (int r = 0; r < 8; ++r) {
                        float v = acc[e8][r] + bias;
                        sPre[eb + e8][(mhi + r) * c + colj] = v > 0.f ? v : 0.f;
                    }
                }
            }
        }
        __syncthreads();
    }

    // ---- out: [16 elems x 16c] @ [16c x c], A fragment shared across N tiles ----
    const int L = 16 * c;                      // K length, always multiple of 4
    {
        v8f zero = {};
        v8f oacc[CT];
        #pragma unroll
        for (int nt = 0; nt < CT; ++nt) oacc[nt] = zero;

        for (int k = 0; k < (L >> 2); ++k) {
            int k0 = 4 * k + khi, k1 = k0 + 1;
            v2f a;                              // one A fragment for all N tiles
            a.x = sPre[row][k0];
            a.y = sPre[row][k1];
            #pragma unroll
            for (int nt = 0; nt < CT; ++nt) {
                int colj = nt * 16 + row;
                v2f bb;
                bb.x = (colj < c) ? outW[k0 * c + colj] : 0.f;
                bb.y = (colj < c) ? outW[k1 * c + colj] : 0.f;
                oacc[nt] = __builtin_amdgcn_wmma_f32_16x16x4_f32(false, a, false, bb,
                                                                 (short)0, oacc[nt],
                                                                 false, false);
            }
        }
        #pragma unroll
        for (int nt = 0; nt < CT; ++nt) {
            int colj = nt * 16 + row;
            if (colj < c) {
                float bias = outB[colj];
                #pragma unroll
                for (int r = 0; r < 8; ++r) {
                    float v = oacc[nt][r] + bias;
                    alout[(size_t)(b * Eq + e0 + mhi + r) * c + colj] = v > 0.f ? v : 0.f;
                }
            }
        }
    }
}

// masked mean pool over E (final layer has c = 40)
__global__ __launch_bounds__(64)
void pool_kernel(const float* __restrict__ al, const int* __restrict__ nume,
                 float* __restrict__ pooled)
{
    int b = blockIdx.x;
    int j = threadIdx.x;
    if (j >= 40) return;
    int n = nume[b];
    if (n < 1) n = 1;
    float s = 0.f;
    for (int e = 0; e < n; ++e) s += al[(b * Eq + e) * 40 + j];
    pooled[b * 40 + j] = s / (float)n;
}

// tiny MLP head: 40 -> 100 -> 100 -> 6 on 32 rows
__global__ __launch_bounds__(128)
void fc_kernel(const float* __restrict__ pooled,
               const float* __restrict__ fc1W, const float* __restrict__ fc1B,
               const float* __restrict__ fc2W, const float* __restrict__ fc2B,
               const float* __restrict__ fc3W, const float* __restrict__ fc3B,
               float* __restrict__ out)
{
    __shared__ float sp[Bq * 40];
    __shared__ float h1[Bq * 100];
    __shared__ float h2[Bq * 100];
    int t = threadIdx.x;
    for (int i = t; i < Bq * 40; i += 128) sp[i] = pooled[i];
    __syncthreads();
    for (int i = t; i < Bq * 100; i += 128) {
        int r = i / 100, cc = i - r * 100;
        float s = fc1B[cc];
        for (int k = 0; k < 40; ++k) s += sp[r * 40 + k] * fc1W[k * 100 + cc];
        h1[i] = s > 0.f ? s : 0.f;
    }
    __syncthreads();
    for (int i = t; i < Bq * 100; i += 128) {
        int r = i / 100, cc = i - r * 100;
        float s = fc2B[cc];
        for (int k = 0; k < 100; ++k) s += h1[r * 100 + k] * fc2W[k * 100 + cc];
        h2[i] = s > 0.f ? s : 0.f;
    }
    __syncthreads();
    for (int i = t; i < Bq * 6; i += 128) {
        int r = i / 6, cc = i - r * 6;
        float s = fc3B[cc];
        for (int k = 0; k < 100; ++k) s += h2[r * 100 + k] * fc3W[k * 6 + cc];
        out[i] = s;
    }
}

extern "C" void kernel_launch(void* const* d_in, const int* in_sizes, int n_in,
                              void* d_out, int out_size, void* d_ws, size_t ws_size,
                              hipStream_t stream)
{
    (void)in_sizes; (void)n_in; (void)out_size; (void)ws_size;

    const float* space = (const float*)d_in[0];   // [32][4096][4]
    const float* allf  = (const float*)d_in[1];   // [32][4096][16]
    const int*   nbr   = (const int*)d_in[2];     // [32][4096][16]
    const int*   nume  = (const int*)d_in[3];     // [32][1]

    // params: pytree flatten (sorted dict keys):
    // conv[0..5]{out{W,b},pre{W,b},wfac{W,b}}, fc1{W,b}, fc2{W,b}, fc3{W,b}
    const float* outWp[6]; const float* outBp[6];
    const float* preWp[6]; const float* preBp[6];
    const float* wfWp[6];  const float* wfBp[6];
    for (int i = 0; i < 6; ++i) {
        int base = 4 + i * 6;
        outWp[i] = (const float*)d_in[base + 0];
        outBp[i] = (const float*)d_in[base + 1];
        preWp[i] = (const float*)d_in[base + 2];
        preBp[i] = (const float*)d_in[base + 3];
        wfWp[i]  = (const float*)d_in[base + 4];
        wfBp[i]  = (const float*)d_in[base + 5];
    }
    const float* fc1W = (const float*)d_in[40];
    const float* fc1B = (const float*)d_in[41];
    const float* fc2W = (const float*)d_in[42];
    const float* fc2B = (const float*)d_in[43];
    const float* fc3W = (const float*)d_in[44];
    const float* fc3B = (const float*)d_in[45];

    // workspace carve-up: two ping-pong feature buffers (max c = 40) + pooled
    float* ws     = (float*)d_ws;
    float* alA    = ws;
    float* alB    = ws + (size_t)Bq * Eq * 40;
    float* pooled = ws + (size_t)2 * Bq * Eq * 40;

    const int fins[6] = {16, 15, 20, 25, 30, 35};
    const int cs[6]   = {15, 20, 25, 30, 35, 40};

    const float* cur = allf;
    dim3 grid(Bq * (Eq / 16));
    for (int l = 0; l < 6; ++l) {
        float* dst = (l & 1) ? alB : alA;
        int ct = (cs[l] + 15) >> 4;
        switch (ct) {
        case 1:
            sconv_layer<1><<<grid, 32, 0, stream>>>(space, nbr, cur, dst,
                wfWp[l], wfBp[l], preWp[l], preBp[l], outWp[l], outBp[l],
                fins[l], cs[l]);
            break;
        case 2:
            sconv_layer<2><<<grid, 32, 0, stream>>>(space, nbr, cur, dst,
                wfWp[l], wfBp[l], preWp[l], preBp[l], outWp[l], outBp[l],
                fins[l], cs[l]);
            break;
        default:
            sconv_layer<3><<<grid, 32, 0, stream>>>(space, nbr, cur, dst,
                wfWp[l], wfBp[l], preWp[l], preBp[l], outWp[l], outBp[l],
                fins[l], cs[l]);
            break;
        }
        cur = dst;
    }
    // after 6 layers, result is in alB (c = 40)
    pool_kernel<<<Bq, 64, 0, stream>>>(alB, nume, pooled);
    fc_kernel<<<1, 128, 0, stream>>>(pooled, fc1W, fc1B, fc2W, fc2B, fc3W, fc3B,
                                     (float*)d_out);
}